// GatedAttnLayer_25512105738337
// MI455X (gfx1250) — compile-verified
//
#include <hip/hip_runtime.h>
#include <math.h>

#define N_NODES 10000
#define N_EDGES 320000
#define NHEAD 4
#define NDIM 64
#define NCLS 16
#define NIN 256
#define NHD (NHEAD*NDIM)            // 256
#define TOT_NH (N_NODES*NHEAD)      // 40000
#define TOT_NHD (N_NODES*NHD)       // 2560000
#define M_TILES (N_NODES/16)        // 625

typedef __attribute__((ext_vector_type(2))) float v2f;
typedef __attribute__((ext_vector_type(8))) float v8f;

// ---------------------------------------------------------------------------
// feat = h @ Wfc  via V_WMMA_F32_16X16X4_F32 (full f32 precision, matches ref)
// 8 waves/block; block owns one 16-col slice of Wfc staged in LDS (padded).
// ---------------------------------------------------------------------------
__global__ __launch_bounds__(256) void k_gemm_feat(const float* __restrict__ h,
                                                   const float* __restrict__ W,
                                                   float* __restrict__ feat) {
  __shared__ float Bl[16 * 258];           // col-major, +2 pad vs 256 to spread banks
  const int tid   = threadIdx.x;
  const int lane  = tid & 31;
  const int wv    = tid >> 5;
  const int tileN = blockIdx.x & 15;
  const int tileM = (blockIdx.x >> 4) * 8 + wv;
  const int col0  = tileN * 16;

  for (int idx = tid; idx < 256 * 16; idx += 256) {
    const int k = idx >> 4, c = idx & 15;
    Bl[c * 258 + k] = W[k * NHD + col0 + c];
  }
  __syncthreads();

  if (tileM >= M_TILES) return;            // whole-wave exit; EXEC stays all-1 for WMMA

  const int half = lane >> 4;              // K sub-pair select per ISA A/B layout
  const int l15  = lane & 15;
  const float* ap = h + (size_t)(tileM * 16 + l15) * NIN + half * 2;
  const float* bp = &Bl[l15 * 258 + half * 2];

  v8f acc = {0.f, 0.f, 0.f, 0.f, 0.f, 0.f, 0.f, 0.f};
#pragma unroll 8
  for (int k = 0; k < NIN; k += 4) {
    v2f a = *(const v2f*)(ap + k);         // A[M=l15][k+2*half .. +1]
    v2f b = *(const v2f*)(bp + k);         // B[k+2*half .. +1][N=l15]
    acc = __builtin_amdgcn_wmma_f32_16x16x4_f32(false, a, false, b,
                                                (short)0, acc, false, false);
  }

  // C/D layout: lanes 0-15 hold M=v, lanes 16-31 hold M=8+v, col = lane&15
  float* op = feat + (size_t)(tileM * 16 + half * 8) * NHD + col0 + l15;
#pragma unroll
  for (int v = 0; v < 8; ++v) op[v * NHD] = acc[v];
}

// ---------------------------------------------------------------------------
__global__ void k_init_emax(float* emax) {
  int i = blockIdx.x * blockDim.x + threadIdx.x;
  if (i < TOT_NH) emax[i] = -INFINITY;
}

__global__ void k_pred(const float* __restrict__ logits, int* __restrict__ pred) {
  int n = blockIdx.x * blockDim.x + threadIdx.x;
  if (n >= N_NODES) return;
  const float* row = logits + (size_t)n * NCLS;
  float best = row[0]; int bi = 0;
#pragma unroll
  for (int c = 1; c < NCLS; ++c) { float v = row[c]; if (v > best) { best = v; bi = c; } }
  pred[n] = bi;
}

__global__ void k_degs(const int* __restrict__ dst, float* __restrict__ degs) {
  int e = blockIdx.x * blockDim.x + threadIdx.x;
  if (e < N_EDGES) atomicAdd(&degs[dst[e]], 1.0f);
}

__global__ void k_elr(const float* __restrict__ feat, const float* __restrict__ al,
                      const float* __restrict__ ar, float* __restrict__ el,
                      float* __restrict__ er) {
  int i = blockIdx.x * blockDim.x + threadIdx.x;   // over N*H
  if (i >= TOT_NH) return;
  int hh = i & (NHEAD - 1);
  const float* f = feat + (size_t)i * NDIM;        // n*256 + h*64 == i*64
  const float* L = al + hh * NDIM;
  const float* R = ar + hh * NDIM;
  float sl = 0.f, sr = 0.f;
  for (int d = 0; d < NDIM; ++d) { float v = f[d]; sl += v * L[d]; sr += v * R[d]; }
  el[i] = sl; er[i] = sr;
}

__device__ __forceinline__ float leaky02(float x) { return x > 0.f ? x : 0.2f * x; }

__device__ __forceinline__ void atomicMaxF(float* addr, float val) {
  if (val >= 0.f) atomicMax((int*)addr, __float_as_int(val));
  else            atomicMin((unsigned int*)addr, __float_as_uint(val));
}

__global__ void k_edge_max(const int* __restrict__ src, const int* __restrict__ dst,
                           const float* __restrict__ el, const float* __restrict__ er,
                           float* __restrict__ emax) {
  int e = blockIdx.x * blockDim.x + threadIdx.x;
  if (e >= N_EDGES) return;
  int s = src[e], d = dst[e];
  float4 l4 = *(const float4*)(el + (size_t)s * NHEAD);
  float4 r4 = *(const float4*)(er + (size_t)d * NHEAD);
  atomicMaxF(&emax[d * NHEAD + 0], leaky02(l4.x + r4.x));
  atomicMaxF(&emax[d * NHEAD + 1], leaky02(l4.y + r4.y));
  atomicMaxF(&emax[d * NHEAD + 2], leaky02(l4.z + r4.z));
  atomicMaxF(&emax[d * NHEAD + 3], leaky02(l4.w + r4.w));
}

__global__ void k_edge_exp(const int* __restrict__ src, const int* __restrict__ dst,
                           const float* __restrict__ el, const float* __restrict__ er,
                           const float* __restrict__ emax, float* __restrict__ eexp,
                           float* __restrict__ denom) {
  int e = blockIdx.x * blockDim.x + threadIdx.x;
  if (e >= N_EDGES) return;
  int s = src[e], d = dst[e];
  float4 l4 = *(const float4*)(el + (size_t)s * NHEAD);
  float4 r4 = *(const float4*)(er + (size_t)d * NHEAD);
  float4 m4 = *(const float4*)(emax + (size_t)d * NHEAD);
  float x0 = __expf(leaky02(l4.x + r4.x) - m4.x);
  float x1 = __expf(leaky02(l4.y + r4.y) - m4.y);
  float x2 = __expf(leaky02(l4.z + r4.z) - m4.z);
  float x3 = __expf(leaky02(l4.w + r4.w) - m4.w);
  *(float4*)(eexp + (size_t)e * NHEAD) = make_float4(x0, x1, x2, x3);
  atomicAdd(&denom[d * NHEAD + 0], x0);
  atomicAdd(&denom[d * NHEAD + 1], x1);
  atomicAdd(&denom[d * NHEAD + 2], x2);
  atomicAdd(&denom[d * NHEAD + 3], x3);
}

// One wave32 per (edge, head): coalesced 128B gather of feat[src], L2 atomics out.
__global__ __launch_bounds__(256) void k_edge_scatter(
    const int* __restrict__ src, const int* __restrict__ dst,
    const float* __restrict__ eexp, const float* __restrict__ denom,
    const int* __restrict__ pred, const float* __restrict__ feat,
    float* __restrict__ f1, float* __restrict__ cnts, int* __restrict__ present,
    float* __restrict__ agg) {
  int gw   = (blockIdx.x * blockDim.x + threadIdx.x) >> 5;
  int lane = threadIdx.x & 31;
  if (gw >= N_EDGES * NHEAD) return;
  int e = gw >> 2, hh = gw & 3;
  int s = src[e], d = dst[e];
  float a = eexp[(size_t)e * NHEAD + hh] / denom[d * NHEAD + hh];
  if (lane == 0) {
    int ps = pred[s];
    if (ps == pred[d]) atomicAdd(&f1[d * NHEAD + hh], a);
    atomicAdd(&cnts[(size_t)(d * NHEAD + hh) * NCLS + ps], a);
    present[ps] = 1;
  }
  const float* fs = feat + (size_t)s * NHD + hh * NDIM;
  float*       ag = agg  + (size_t)d * NHD + hh * NDIM;
  atomicAdd(&ag[lane],      fs[lane]      * a);
  atomicAdd(&ag[lane + 32], fs[lane + 32] * a);
}

// Per-(n,h): finalize f1, compute entropy f2, block-reduce LayerNorm statistics.
__global__ __launch_bounds__(256) void k_node_feats(
    const float* __restrict__ degs, const float* __restrict__ cnts,
    const int* __restrict__ present, float* __restrict__ f1,
    float* __restrict__ f2, float* __restrict__ red) {
  __shared__ float4 sh[256];
  int i = blockIdx.x * 256 + threadIdx.x;
  float f1v = 0.f, f2v = 0.f;
  if (i < TOT_NH) {
    int n = i >> 2;
    float degc = fmaxf(degs[n], 1.0f);
    f1v = f1[i] / degc;
    const float* cr = cnts + (size_t)i * NCLS;
    float s = 0.f;
#pragma unroll
    for (int c = 0; c < NCLS; ++c) {
      if (present[c]) {
        float p = fmaxf(cr[c] / degc, 1e-5f);
        s -= p * __logf(p);
      }
    }
    f2v = s;
    f1[i] = f1v;
    f2[i] = f2v;
  }
  sh[threadIdx.x] = make_float4(f1v, f1v * f1v, f2v, f2v * f2v);
  __syncthreads();
  for (int off = 128; off > 0; off >>= 1) {
    if ((int)threadIdx.x < off) {
      float4 a = sh[threadIdx.x], b = sh[threadIdx.x + off];
      sh[threadIdx.x] = make_float4(a.x + b.x, a.y + b.y, a.z + b.z, a.w + b.w);
    }
    __syncthreads();
  }
  if (threadIdx.x == 0) {
    atomicAdd(&red[0], sh[0].x); atomicAdd(&red[1], sh[0].y);
    atomicAdd(&red[2], sh[0].z); atomicAdd(&red[3], sh[0].w);
  }
}

__device__ __forceinline__ float sigm(float x) { return 1.f / (1.f + __expf(-x)); }

__global__ void k_gate_z(const float* __restrict__ f1, const float* __restrict__ f2,
                         const float* __restrict__ red, const float* __restrict__ old_z,
                         const float* __restrict__ tau1, const float* __restrict__ tau2,
                         float* __restrict__ zout, float* __restrict__ gate) {
  int i = blockIdx.x * blockDim.x + threadIdx.x;
  if (i >= TOT_NH) return;
  const float M = (float)TOT_NH;
  float mu1 = red[0] / M, var1 = red[1] / M - mu1 * mu1;
  float mu2 = red[2] / M, var2 = red[3] / M - mu2 * mu2;
  float nf1 = (f1[i] - mu1) * rsqrtf(var1 + 1e-5f);
  float nf2 = (f2[i] - mu2) * rsqrtf(var2 + 1e-5f);
  float z = sigm(tau1[0] - nf1) * sigm(tau2[0] - nf2);
  zout[i] = z;
  gate[i] = fminf(old_z[i], z);
}

__global__ void k_final(const float* __restrict__ feat, const float* __restrict__ agg,
                        const float* __restrict__ gate, const float* __restrict__ degs,
                        float* __restrict__ out) {
  int i = blockIdx.x * blockDim.x + threadIdx.x;
  if (i >= TOT_NHD) return;
  int nh = i >> 6;                // n*4 + h
  int n  = i >> 8;
  float nrm = rsqrtf(fmaxf(degs[n], 1.0f));
  out[i] = feat[i] + gate[nh] * nrm * agg[i];
}

// ---------------------------------------------------------------------------
extern "C" void kernel_launch(void* const* d_in, const int* in_sizes, int n_in,
                              void* d_out, int out_size, void* d_ws, size_t ws_size,
                              hipStream_t stream) {
  (void)in_sizes; (void)n_in; (void)out_size; (void)ws_size;
  const float* h      = (const float*)d_in[0];
  const float* logits = (const float*)d_in[1];
  const float* old_z  = (const float*)d_in[2];
  const float* attn_l = (const float*)d_in[3];
  const float* attn_r = (const float*)d_in[4];
  const float* Wfc    = (const float*)d_in[5];
  const float* tau1   = (const float*)d_in[6];
  const float* tau2   = (const float*)d_in[7];
  const int*   src    = (const int*)d_in[8];
  const int*   dst    = (const int*)d_in[9];

  float* feat  = (float*)d_ws;
  float* eexp  = feat  + TOT_NHD;
  float* elv   = eexp  + (size_t)N_EDGES * NHEAD;
  float* erv   = elv   + TOT_NH;
  float* emax  = erv   + TOT_NH;
  float* f2    = emax  + TOT_NH;
  // ---- contiguous zero block begins here ----
  float* agg   = f2    + TOT_NH;
  float* denom = agg   + TOT_NHD;
  float* f1    = denom + TOT_NH;
  float* degs  = f1    + TOT_NH;
  float* red   = degs  + N_NODES;
  float* cnts  = red   + 8;
  int*   present = (int*)(cnts + (size_t)N_NODES * NHEAD * NCLS);
  // ---- contiguous zero block ends here ----
  int*   pred    = present + NCLS;

  const size_t zero_bytes =
      ((size_t)TOT_NHD + TOT_NH + TOT_NH + N_NODES + 8 +
       (size_t)N_NODES * NHEAD * NCLS + NCLS) * sizeof(float);
  hipMemsetAsync(agg, 0, zero_bytes, stream);

  float* out_newh = (float*)d_out;
  float* out_z    = out_newh + TOT_NHD;

  k_init_emax<<<(TOT_NH + 255) / 256, 256, 0, stream>>>(emax);
  k_gemm_feat<<<((M_TILES + 7) / 8) * 16, 256, 0, stream>>>(h, Wfc, feat);
  k_pred<<<(N_NODES + 255) / 256, 256, 0, stream>>>(logits, pred);
  k_degs<<<(N_EDGES + 255) / 256, 256, 0, stream>>>(dst, degs);
  k_elr<<<(TOT_NH + 255) / 256, 256, 0, stream>>>(feat, attn_l, attn_r, elv, erv);
  k_edge_max<<<(N_EDGES + 255) / 256, 256, 0, stream>>>(src, dst, elv, erv, emax);
  k_edge_exp<<<(N_EDGES + 255) / 256, 256, 0, stream>>>(src, dst, elv, erv, emax,
                                                        eexp, denom);
  k_edge_scatter<<<(N_EDGES * NHEAD) / 8, 256, 0, stream>>>(
      src, dst, eexp, denom, pred, feat, f1, cnts, present, agg);
  k_node_feats<<<(TOT_NH + 255) / 256, 256, 0, stream>>>(degs, cnts, present,
                                                         f1, f2, red);
  k_gate_z<<<(TOT_NH + 255) / 256, 256, 0, stream>>>(f1, f2, red, old_z, tau1, tau2,
                                                     out_z, /*gate=*/f1);
  k_final<<<(TOT_NHD + 255) / 256, 256, 0, stream>>>(feat, agg, f1, degs, out_newh);
}